// WeightsLib2D_9165460209858
// MI455X (gfx1250) — compile-verified
//
#include <hip/hip_runtime.h>

typedef float v2f __attribute__((ext_vector_type(2)));
typedef float v4f __attribute__((ext_vector_type(4)));
typedef float v8f __attribute__((ext_vector_type(8)));

namespace {
constexpr int kS0 = 4096;
constexpr int kS1 = 8192;
constexpr int kWaves = 8;                        // waves per block
constexpr int kNJ = 4;                           // 16-wide tiles per wave
constexpr int kBlockCols = 16 * kWaves * kNJ;    // 512 columns per block

// (w + off)^2 * scl, complex
__device__ __forceinline__ void xform(float wr, float wi, float offr, float offi,
                                      float sclr, float scli, float& mr, float& mi_) {
    const float tr = wr + offr;
    const float ti = wi + offi;
    const float sqr = fmaf(tr, tr, -ti * ti);
    const float sqi = 2.0f * tr * ti;
    mr  = fmaf(sqr, sclr, -sqi * scli);
    mi_ = fmaf(sqr, scli,  sqi * sclr);
}
} // namespace

__global__ __launch_bounds__(256)
void wl2d_fused(const float* __restrict__ wb_r, const float* __restrict__ wb_i,
                const float* __restrict__ wmi_r, const float* __restrict__ wmi_i,
                const float* __restrict__ wmj_r, const float* __restrict__ wmj_i,
                const float* __restrict__ bc_r,  const float* __restrict__ bc_i,
                const float* __restrict__ mc_r,  const float* __restrict__ mc_i,
                float* __restrict__ out)
{
    const int lane = threadIdx.x & 31;
    const int wave = threadIdx.x >> 5;
    const int li   = lane & 15;   // tile row (output row) index & mj column index
    const int hi   = lane >> 4;   // half select (K offset / column-half offset)
    const int rb   = hi << 1;

    const int i0    = blockIdx.y << 4;
    const int jbase = blockIdx.x * kBlockCols + wave * 16;

    const float bc0r = bc_r[0], bc0i = bc_i[0];
    const float bc1r = bc_r[1], bc1i = bc_i[1];

    // ---- B operands (mi side): mi[r][i0+li] for r in {rb, rb+1, rb+4, rb+5}
    float mir[4], mii[4];
    {
        const int i = i0 + li;
        #pragma unroll
        for (int t = 0; t < 4; ++t) {
            const int r = rb + (t & 1) + ((t >> 1) << 2);   // rb, rb+1, rb+4, rb+5
            const float wr = wmi_r[r * kS0 + i];
            const float wi = wmi_i[r * kS0 + i];
            xform(wr, wi, mc_r[2 * r], mc_i[2 * r], mc_r[2 * r + 1], mc_i[2 * r + 1],
                  mir[t], mii[t]);
        }
    }

    for (int t = 0; t < kNJ; ++t) {
        const int j0 = jbase + t * (16 * kWaves);
        const int j  = j0 + li;

        // ---- A operands (mj side): mj[r][j] for r in {rb, rb+1, rb+4, rb+5}
        float mjr[4], mji[4], nji[4];
        #pragma unroll
        for (int u = 0; u < 4; ++u) {
            const int r = rb + (u & 1) + ((u >> 1) << 2);
            const float wr = wmj_r[r * kS1 + j];
            const float wi = wmj_i[r * kS1 + j];
            xform(wr, wi, mc_r[16 + 2 * r], mc_i[16 + 2 * r],
                  mc_r[16 + 2 * r + 1], mc_i[16 + 2 * r + 1], mjr[u], mji[u]);
            nji[u] = -mji[u];
        }

        // ---- transposed rank-16 real GEMM pair (D = mod^T tile):
        //      modT_r = [Re(mj); -Im(mj)]^T_tile x [Re(mi); Im(mi)]
        //      modT_i = [Im(mj);  Re(mj)]^T_tile x [Re(mi); Im(mi)]
        v8f cr = {0.f,0.f,0.f,0.f,0.f,0.f,0.f,0.f};
        v8f ci = {0.f,0.f,0.f,0.f,0.f,0.f,0.f,0.f};
        v2f a, b;

        b.x = mir[0]; b.y = mir[1];                 // K = 0..3
        a.x = mjr[0]; a.y = mjr[1];
        cr = __builtin_amdgcn_wmma_f32_16x16x4_f32(false, a, false, b, (short)0, cr, false, false);
        a.x = mji[0]; a.y = mji[1];
        ci = __builtin_amdgcn_wmma_f32_16x16x4_f32(false, a, false, b, (short)0, ci, false, false);

        b.x = mir[2]; b.y = mir[3];                 // K = 4..7
        a.x = mjr[2]; a.y = mjr[3];
        cr = __builtin_amdgcn_wmma_f32_16x16x4_f32(false, a, false, b, (short)0, cr, false, false);
        a.x = mji[2]; a.y = mji[3];
        ci = __builtin_amdgcn_wmma_f32_16x16x4_f32(false, a, false, b, (short)0, ci, false, false);

        b.x = mii[0]; b.y = mii[1];                 // K = 8..11
        a.x = nji[0]; a.y = nji[1];
        cr = __builtin_amdgcn_wmma_f32_16x16x4_f32(false, a, false, b, (short)0, cr, false, false);
        a.x = mjr[0]; a.y = mjr[1];
        ci = __builtin_amdgcn_wmma_f32_16x16x4_f32(false, a, false, b, (short)0, ci, false, false);

        b.x = mii[2]; b.y = mii[3];                 // K = 12..15
        a.x = nji[2]; a.y = nji[3];
        cr = __builtin_amdgcn_wmma_f32_16x16x4_f32(false, a, false, b, (short)0, cr, false, false);
        a.x = mjr[2]; a.y = mjr[3];
        ci = __builtin_amdgcn_wmma_f32_16x16x4_f32(false, a, false, b, (short)0, ci, false, false);

        // ---- streaming epilogue, transposed C/D layout:
        //      lane li <-> output row (i0+li), VGPR v <-> output column (j0 + 8*hi + v)
        //      => each lane owns 8 contiguous floats of one row: 2x b128 per array.
        const long base = (long)(i0 + li) * kS1 + (j0 + 8 * hi);
        const v4f br0 = __builtin_nontemporal_load((const v4f*)(wb_r + base));
        const v4f br1 = __builtin_nontemporal_load((const v4f*)(wb_r + base + 4));
        const v4f bi0 = __builtin_nontemporal_load((const v4f*)(wb_i + base));
        const v4f bi1 = __builtin_nontemporal_load((const v4f*)(wb_i + base + 4));
        v4f o0, o1;
        #pragma unroll
        for (int v = 0; v < 4; ++v) {
            {   // columns j0+8hi+v
                const float tr = br0[v] + bc0r;
                const float ti = bi0[v] + bc0i;
                const float sqr = fmaf(tr, tr, -ti * ti);
                const float sqi = 2.0f * tr * ti;
                const float baser = fmaf(sqr, bc1r, -sqi * bc1i);
                const float basei = fmaf(sqr, bc1i,  sqi * bc1r);
                o0[v] = fmaf(baser, cr[v], -basei * ci[v]);
            }
            {   // columns j0+8hi+4+v
                const float tr = br1[v] + bc0r;
                const float ti = bi1[v] + bc0i;
                const float sqr = fmaf(tr, tr, -ti * ti);
                const float sqi = 2.0f * tr * ti;
                const float baser = fmaf(sqr, bc1r, -sqi * bc1i);
                const float basei = fmaf(sqr, bc1i,  sqi * bc1r);
                o1[v] = fmaf(baser, cr[v + 4], -basei * ci[v + 4]);
            }
        }
        __builtin_nontemporal_store(o0, (v4f*)(out + base));
        __builtin_nontemporal_store(o1, (v4f*)(out + base + 4));
    }
}

extern "C" void kernel_launch(void* const* d_in, const int* in_sizes, int n_in,
                              void* d_out, int out_size, void* d_ws, size_t ws_size,
                              hipStream_t stream) {
    (void)in_sizes; (void)n_in; (void)d_ws; (void)ws_size; (void)out_size;
    const float* wb_r  = (const float*)d_in[0];
    const float* wb_i  = (const float*)d_in[1];
    const float* wmi_r = (const float*)d_in[2];
    const float* wmi_i = (const float*)d_in[3];
    const float* wmj_r = (const float*)d_in[4];
    const float* wmj_i = (const float*)d_in[5];
    const float* bc_r  = (const float*)d_in[6];
    const float* bc_i  = (const float*)d_in[7];
    const float* mc_r  = (const float*)d_in[8];
    const float* mc_i  = (const float*)d_in[9];
    float* out = (float*)d_out;

    dim3 grid(kS1 / kBlockCols, kS0 / 16);   // (16, 256)
    wl2d_fused<<<grid, 256, 0, stream>>>(wb_r, wb_i, wmi_r, wmi_i, wmj_r, wmj_i,
                                         bc_r, bc_i, mc_r, mc_i, out);
}